// LCGNoiseLayer2D_25898652795149
// MI455X (gfx1250) — compile-verified
//
#include <hip/hip_runtime.h>
#include <hip/hip_bf16.h>

typedef __attribute__((ext_vector_type(2))) float v2f;
typedef __attribute__((ext_vector_type(8))) float v8f;

#define B_SZ  32
#define CIN   256
#define COUT  512
#define HW    4096
#define M_TOT (B_SZ * HW)                 // 131072 pixels
#define NOISE_SCALE 2.3283064370807974e-11f   // 0.1 / 4294967295.0
#define BN_EPS 1e-5f
#define SEED_0 12345u

// ---- compile-time LCG jump-ahead table: F_{2^i}(s) = p[i]*s + q[i]  (mod 2^32)
// one step: s -> 65539*s + 1 ; value_t uses seed after (t+1) steps.
struct LCGTab { unsigned p[16]; unsigned q[16]; };
constexpr LCGTab makeTab() {
  LCGTab t{};
  unsigned p = 65539u, q = 1u;                 // F_{2^0}
  for (int i = 0; i < 16; ++i) {
    t.p[i] = p; t.q[i] = q;
    q = p * q + q;                             // g(2n) = g(n)*(M^n + 1)
    p = p * p;                                 // M^(2n)
  }
  return t;
}
__constant__ LCGTab LCG = makeTab();

// =====================================================================
// Kernel 1: fused noise + ReLU + 1x1-conv GEMM via V_WMMA_F32_16X16X4_F32
//   D[o, m] = sum_c W[o,c] * relu(x[c,m] + noise(c, hw))
//   wave tile: 16 m  x 64 o ; workgroup: 8 waves = 128 m x 64 o
// =====================================================================
__global__ __launch_bounds__(256) void gemm_kernel(
    const float* __restrict__ x, const float* __restrict__ w,
    float* __restrict__ y) {
  __shared__ float wlds[64 * 68];              // 64 o-rows x 64 c-slab (pad 68 -> conflict-free)

  const int tid   = threadIdx.x;
  const int lane  = tid & 31;
  const int lhalf = lane & 15;
  const int wave  = tid >> 5;
  const int koff  = (lane >> 4) << 1;          // lanes 0-15: K+0/K+1 ; lanes 16-31: K+2/K+3

  const int m0 = (blockIdx.x * 8 + wave) * 16; // 16-aligned -> whole tile in one batch image
  const int o0 = blockIdx.y * 64;

  const int m  = m0 + lhalf;                   // pixel = b*4096 + hw
  const int b  = m >> 12;
  const int hw = m & 4095;

  // x element at (c, m): x[b*CIN*HW + c*HW + hw]
  const float* px = x + (size_t)b * CIN * HW + hw + (size_t)koff * HW;

  // ---- per-lane LCG seed for t = c*4096 + hw, starting at c = koff
  unsigned P = 1u, Q = 0u;
  const unsigned e = (unsigned)hw + 1u;        // seed(t=hw) = F_{hw+1}(SEED)
  #pragma unroll
  for (int i = 0; i < 13; ++i) {
    if (e & (1u << i)) { Q = LCG.p[i] * Q + LCG.q[i]; P = LCG.p[i] * P; }
  }
  unsigned s1 = P * SEED_0 + Q;                // c = 0
  if (lane >= 16) s1 = LCG.p[13] * s1 + LCG.q[13];     // jump Δt = 2*4096 (c += 2)
  unsigned s2 = LCG.p[12] * s1 + LCG.q[12];            // jump Δt = 4096   (c += 1)
  const unsigned stepP = LCG.p[14], stepQ = LCG.q[14]; // per-iter jump Δt = 4*4096

  v8f acc0 = {}, acc1 = {}, acc2 = {}, acc3 = {};

  for (int ks = 0; ks < 4; ++ks) {             // K slabs of 64
    __syncthreads();
    {   // cooperative weight slab load: conv_w[o0+ol, ks*64 .. +64)
      const int ol = tid >> 2;
      const int cq = (tid & 3) * 16;
      const float4* src = (const float4*)(w + (size_t)(o0 + ol) * CIN + ks * 64 + cq);
      float4* dst = (float4*)(&wlds[ol * 68 + cq]);
      dst[0] = src[0]; dst[1] = src[1]; dst[2] = src[2]; dst[3] = src[3];
    }
    __syncthreads();

    #pragma unroll 4
    for (int kk = 0; kk < 64; kk += 4) {
      // B-frag (X2^T, 4c x 16m): vgpr0 = k+koff, vgpr1 = k+koff+1
      const float x0 = px[0];
      const float x1 = px[HW];
      px += 4 * HW;
      const float n0 = (float)s1 * NOISE_SCALE;
      const float n1 = (float)s2 * NOISE_SCALE;
      s1 = stepP * s1 + stepQ;
      s2 = stepP * s2 + stepQ;
      v2f bf;
      bf.x = fmaxf(x0 + n0, 0.0f);
      bf.y = fmaxf(x1 + n1, 0.0f);

      // A-frags (W, 16o x 4c) from LDS: lane = o-row, vgpr0/1 = k+koff / +1
      const int ci = kk + koff;
      const v2f a0 = *(const v2f*)(&wlds[(0 * 16 + lhalf) * 68 + ci]);
      const v2f a1 = *(const v2f*)(&wlds[(1 * 16 + lhalf) * 68 + ci]);
      const v2f a2 = *(const v2f*)(&wlds[(2 * 16 + lhalf) * 68 + ci]);
      const v2f a3 = *(const v2f*)(&wlds[(3 * 16 + lhalf) * 68 + ci]);

      acc0 = __builtin_amdgcn_wmma_f32_16x16x4_f32(false, a0, false, bf, (short)0, acc0, false, false);
      acc1 = __builtin_amdgcn_wmma_f32_16x16x4_f32(false, a1, false, bf, (short)0, acc1, false, false);
      acc2 = __builtin_amdgcn_wmma_f32_16x16x4_f32(false, a2, false, bf, (short)0, acc2, false, false);
      acc3 = __builtin_amdgcn_wmma_f32_16x16x4_f32(false, a3, false, bf, (short)0, acc3, false, false);
    }
  }

  // ---- store D: lane = pixel column (coalesced), vgpr r = o-row r (+8 for hi half)
  // conv_b is provably cancelled by BatchNorm (constant per channel) -> skipped.
  const int rowAdd = (lane >> 4) << 3;
  float* py = y + ((size_t)b * COUT + o0) * HW + hw;
  #pragma unroll
  for (int nt = 0; nt < 4; ++nt) {
    const v8f a = nt == 0 ? acc0 : nt == 1 ? acc1 : nt == 2 ? acc2 : acc3;
    #pragma unroll
    for (int r = 0; r < 8; ++r) {
      const int o_loc = nt * 16 + r + rowAdd;
      py[(size_t)o_loc * HW] = a[r];
    }
  }
}

// =====================================================================
// Kernel 0: zero the stats accumulators in workspace
// =====================================================================
__global__ void zero_kernel(float* __restrict__ ws) {
  ws[blockIdx.x * blockDim.x + threadIdx.x] = 0.0f;
}

// =====================================================================
// Kernel 2: per-channel sum / sum-of-squares over (B, H, W), coalesced
// =====================================================================
__global__ __launch_bounds__(256) void stats_kernel(
    const float* __restrict__ y, float* __restrict__ ws) {
  __shared__ float ssum[256];
  __shared__ float ssq[256];
  const int o  = blockIdx.x;
  const int b4 = blockIdx.y * 4;
  float s = 0.0f, q = 0.0f;
  #pragma unroll
  for (int bi = 0; bi < 4; ++bi) {
    const float4* row = (const float4*)(y + ((size_t)(b4 + bi) * COUT + o) * HW);
    #pragma unroll
    for (int j = 0; j < 4; ++j) {
      const float4 v = row[threadIdx.x + j * 256];
      s += v.x + v.y + v.z + v.w;
      q += v.x * v.x + v.y * v.y + v.z * v.z + v.w * v.w;
    }
  }
  ssum[threadIdx.x] = s;
  ssq[threadIdx.x]  = q;
  __syncthreads();
  for (int stride = 128; stride > 0; stride >>= 1) {
    if (threadIdx.x < stride) {
      ssum[threadIdx.x] += ssum[threadIdx.x + stride];
      ssq[threadIdx.x]  += ssq[threadIdx.x + stride];
    }
    __syncthreads();
  }
  if (threadIdx.x == 0) {
    atomicAdd(&ws[o], ssum[0]);
    atomicAdd(&ws[COUT + o], ssq[0]);
  }
}

// =====================================================================
// Kernel 3: mean/var -> per-channel scale & shift
// =====================================================================
__global__ void finalize_kernel(const float* __restrict__ gamma,
                                const float* __restrict__ beta,
                                float* __restrict__ ws) {
  const int o = blockIdx.x * blockDim.x + threadIdx.x;
  if (o < COUT) {
    const float invN = 1.0f / (float)M_TOT;
    const float mean = ws[o] * invN;
    const float var  = ws[COUT + o] * invN - mean * mean;
    const float sc   = gamma[o] * rsqrtf(var + BN_EPS);
    ws[2 * COUT + o] = sc;
    ws[3 * COUT + o] = beta[o] - mean * sc;
  }
}

// =====================================================================
// Kernel 4: in-place normalize (float4, 1024 float4s per (b,o) plane)
// =====================================================================
__global__ __launch_bounds__(256) void norm_kernel(
    float* __restrict__ y, const float* __restrict__ ws) {
  const size_t i4 = (size_t)blockIdx.x * 256 + threadIdx.x;
  const int o = (int)((i4 >> 10) & (COUT - 1));
  const float sc = ws[2 * COUT + o];
  const float sh = ws[3 * COUT + o];
  float4* p = (float4*)y;
  float4 v = p[i4];
  v.x = v.x * sc + sh;
  v.y = v.y * sc + sh;
  v.z = v.z * sc + sh;
  v.w = v.w * sc + sh;
  p[i4] = v;
}

extern "C" void kernel_launch(void* const* d_in, const int* in_sizes, int n_in,
                              void* d_out, int out_size, void* d_ws, size_t ws_size,
                              hipStream_t stream) {
  const float* x      = (const float*)d_in[0];
  const float* conv_w = (const float*)d_in[1];
  const float* conv_b = (const float*)d_in[2];   // cancels under BatchNorm
  const float* gamma  = (const float*)d_in[3];
  const float* beta   = (const float*)d_in[4];
  (void)conv_b; (void)in_sizes; (void)n_in; (void)out_size; (void)ws_size;

  float* y  = (float*)d_out;
  float* ws = (float*)d_ws;

  zero_kernel<<<2, 512, 0, stream>>>(ws);                       // ws[0..1023] = 0

  dim3 ggrid(M_TOT / 128, COUT / 64);                           // 1024 x 8
  gemm_kernel<<<ggrid, 256, 0, stream>>>(x, conv_w, y);

  dim3 sgrid(COUT, B_SZ / 4);                                   // 512 x 8
  stats_kernel<<<sgrid, 256, 0, stream>>>(y, ws);

  finalize_kernel<<<2, 256, 0, stream>>>(gamma, beta, ws);

  const int n4 = (int)(((size_t)B_SZ * COUT * HW) / 4 / 256);   // 65536 blocks
  norm_kernel<<<n4, 256, 0, stream>>>(y, ws);
}